// Median_filter_23648089932319
// MI455X (gfx1250) — compile-verified
//
#include <hip/hip_runtime.h>
#include <hip/hip_bf16.h>

// 3x3 median blur, fp32, zero padding. (B,C,H,W) = (16,3,512,512).
// Bandwidth-bound: ~100 MB total traffic -> ~4.3 us floor at 23.3 TB/s.
// Strategy: stage a 130x34 haloed fp32 tile per workgroup in LDS via
// gfx1250 async global->LDS loads (ASYNCcnt), zero-fill the out-of-image
// halo with ds stores, then each thread computes a 16-tall output column
// with a rolling 3-row window and a 19-exchange median-of-9 network.
// Tile 128x32: halo overhead 4420/4096 = 1.079x of minimal global reads.

#define IMG    512
#define PLANE  (IMG * IMG)
#define TILE_W 128
#define TILE_H 32
#define ROWS_PER_THREAD (TILE_H / 2)  // 256 threads, 128 columns -> 2 row groups
#define HALO_W (TILE_W + 2)          // 130
#define HALO_H (TILE_H + 2)          // 34
#define HALO_N (HALO_W * HALO_H)     // 4420 floats = 17680 B LDS
#define NTHREADS 256

#if defined(__gfx1250__) && __has_builtin(__builtin_amdgcn_global_load_async_to_lds_b32)
#define USE_ASYNC_LDS 1
#else
#define USE_ASYNC_LDS 0
#endif

// Builtin pointer params are (addrspace(1) int*, addrspace(3) int*).
#define GLOBAL_AS_I32(p) ((__attribute__((address_space(1))) int*)(p))
#define LDS_AS_I32(p)    ((__attribute__((address_space(3))) int*)(p))

__device__ __forceinline__ void s2(float& a, float& b) {
    float t = fminf(a, b);
    b = fmaxf(a, b);
    a = t;
}

// Classic 19-compare-exchange median-of-9 (Paeth/Devillard network).
__device__ __forceinline__ float med9(float p0, float p1, float p2,
                                      float p3, float p4, float p5,
                                      float p6, float p7, float p8) {
    s2(p1, p2); s2(p4, p5); s2(p7, p8);
    s2(p0, p1); s2(p3, p4); s2(p6, p7);
    s2(p1, p2); s2(p4, p5); s2(p7, p8);
    s2(p0, p3); s2(p5, p8); s2(p4, p7);
    s2(p3, p6); s2(p1, p4); s2(p2, p5);
    s2(p4, p7); s2(p4, p2); s2(p6, p4);
    s2(p4, p2);
    return p4;
}

__global__ __launch_bounds__(NTHREADS)
void median3x3_kernel(const float* __restrict__ in, float* __restrict__ out) {
    __shared__ float tile[HALO_N];

    const int t  = threadIdx.x;
    const int z  = blockIdx.z;                 // fused (b, c) plane
    const int x0 = blockIdx.x * TILE_W;
    const int y0 = blockIdx.y * TILE_H;

    const float* __restrict__ plane = in + (size_t)z * PLANE;

    // ---- Stage haloed tile into LDS (each HBM byte read ~once) ----
    for (int i = t; i < HALO_N; i += NTHREADS) {
        const int ty = i / HALO_W;
        const int tx = i - ty * HALO_W;
        const int gy = y0 + ty - 1;
        const int gx = x0 + tx - 1;
        const bool inb = (gx >= 0) & (gx < IMG) & (gy >= 0) & (gy < IMG);
#if USE_ASYNC_LDS
        if (inb) {
            // per-lane async DMA: global -> LDS, no VGPR round trip (ASYNCcnt)
            __builtin_amdgcn_global_load_async_to_lds_b32(
                GLOBAL_AS_I32(plane + gy * IMG + gx),
                LDS_AS_I32(&tile[i]),
                /*imm offset*/ 0, /*cpol*/ 0);
        } else {
            tile[i] = 0.0f;   // zero padding for the out-of-image halo
        }
#else
        tile[i] = inb ? plane[gy * IMG + gx] : 0.0f;
#endif
    }

#if USE_ASYNC_LDS
#if __has_builtin(__builtin_amdgcn_s_wait_asynccnt)
    __builtin_amdgcn_s_wait_asynccnt(0);
#else
    asm volatile("s_wait_asynccnt 0" ::: "memory");
#endif
#endif
    __syncthreads();

    // ---- Compute: each thread does a 16-tall column at x = x0+tx ----
    const int tx = t & (TILE_W - 1);            // 0..127
    const int ry = (t >> 7) * ROWS_PER_THREAD;  // 0 or 16: row-group in tile
    const int lx = tx + 1;                      // center column in halo coords

    float a0 = tile[(ry + 0) * HALO_W + lx - 1];
    float a1 = tile[(ry + 0) * HALO_W + lx    ];
    float a2 = tile[(ry + 0) * HALO_W + lx + 1];
    float b0 = tile[(ry + 1) * HALO_W + lx - 1];
    float b1 = tile[(ry + 1) * HALO_W + lx    ];
    float b2 = tile[(ry + 1) * HALO_W + lx + 1];

    float* __restrict__ outp = out + (size_t)z * PLANE;

#pragma unroll
    for (int r = 0; r < ROWS_PER_THREAD; ++r) {
        const int lrow = (ry + r + 2) * HALO_W;
        const float c0 = tile[lrow + lx - 1];
        const float c1 = tile[lrow + lx    ];
        const float c2 = tile[lrow + lx + 1];

        const float m = med9(a0, a1, a2, b0, b1, b2, c0, c1, c2);
        outp[(y0 + ry + r) * IMG + (x0 + tx)] = m;

        a0 = b0; a1 = b1; a2 = b2;   // slide the 3-row window down
        b0 = c0; b1 = c1; b2 = c2;
    }
}

extern "C" void kernel_launch(void* const* d_in, const int* in_sizes, int n_in,
                              void* d_out, int out_size, void* d_ws, size_t ws_size,
                              hipStream_t stream) {
    const float* x = (const float*)d_in[0];
    // d_in[1] is kernel_size (==3 in the reference setup); fixed 3x3 path.
    float* out = (float*)d_out;

    const int planes = in_sizes[0] / PLANE;   // B*C = 48
    dim3 grid(IMG / TILE_W, IMG / TILE_H, planes);
    dim3 block(NTHREADS);
    median3x3_kernel<<<grid, block, 0, stream>>>(x, out);
}